// BA_Shapes_GCN_edge_multiclass_26371099198065
// MI455X (gfx1250) — compile-verified
//
#include <hip/hip_runtime.h>
#include <hip/hip_bf16.h>
#include <math.h>

typedef __attribute__((ext_vector_type(2))) float v2f;
typedef __attribute__((ext_vector_type(8))) float v8f;

#define D 128

// ---------------- degree / norm precompute ----------------

__global__ void k_fill_deg(float* deg, int n) {
  int i = blockIdx.x * blockDim.x + threadIdx.x;
  if (i < n) deg[i] = 1.0f;  // self-loop contributes 1 to every node's degree
}

__global__ void k_count_deg(const int* __restrict__ ei, float* deg, int E) {
  int e = blockIdx.x * blockDim.x + threadIdx.x;
  if (e < E) atomicAdd(&deg[ei[E + e]], 1.0f);  // dst row of edge_index
}

__global__ void k_dinv(const float* __restrict__ deg, float* dinv, int n) {
  int i = blockIdx.x * blockDim.x + threadIdx.x;
  if (i < n) {
    float d = deg[i];
    dinv[i] = d > 0.0f ? rsqrtf(d) : 0.0f;
  }
}

// norm[e] = dinv[src]*dinv[dst]; identical for all 4 layers -> compute once.
__global__ void k_norm(const int* __restrict__ ei, const float* __restrict__ dinv,
                       float* norm, int E, int Etot) {
  int e = blockIdx.x * blockDim.x + threadIdx.x;
  if (e >= Etot) return;
  int s, d;
  if (e < E) { s = ei[e]; d = ei[E + e]; } else { s = d = e - E; }  // self loops appended
  norm[e] = dinv[s] * dinv[d];
}

// ---------------- f32 WMMA GEMM: C[M x 128] = A[M x 128] @ W[128 x 128] ----------------
// One wave32 per 16x16 output tile; 32 steps of V_WMMA_F32_16X16X4_F32 (full f32 precision).

__global__ void k_gemm128(const float* __restrict__ A, const float* __restrict__ W,
                          float* __restrict__ C, int mtiles) {
  int wave = blockIdx.x * (blockDim.x >> 5) + (threadIdx.x >> 5);
  int rt = wave >> 3;        // row tile (16 rows)
  int nt = wave & 7;         // col tile (8 tiles of 16 over 128 cols)
  if (rt >= mtiles) return;  // uniform per-wave; active waves keep EXEC all-ones
  int lane = threadIdx.x & 31;
  int l15  = lane & 15;
  int half = lane >> 4;
  int kb   = half * 2;
  const float* arow = A + (size_t)(rt * 16 + l15) * D;
  int n = nt * 16 + l15;

  v8f c = {};
  #pragma unroll 4
  for (int k0 = 0; k0 < D; k0 += 4) {
    v2f a, b;
    a[0] = arow[k0 + kb];
    a[1] = arow[k0 + kb + 1];
    b[0] = W[(size_t)(k0 + kb) * D + n];
    b[1] = W[(size_t)(k0 + kb + 1) * D + n];
    c = __builtin_amdgcn_wmma_f32_16x16x4_f32(false, a, false, b, (short)0, c,
                                              false, false);
  }
  int rowBase = rt * 16 + half * 8;
  #pragma unroll
  for (int v = 0; v < 8; ++v)
    C[(size_t)(rowBase + v) * D + n] = c[v];
}

// ---------------- bias init / scatter-aggregate / relu ----------------

__global__ void k_init_bias(float* out, const float* __restrict__ b, size_t total) {
  size_t i = blockIdx.x * (size_t)blockDim.x + threadIdx.x;
  if (i < total) out[i] = b[i & (D - 1)];  // out starts at +bias, scatter adds messages
}

// 32 lanes per edge, float4 per lane: out[dst,:] += tmp[src,:] * norm[e]
__global__ void k_scatter(const float* __restrict__ tmp, float* out,
                          const int* __restrict__ ei, const float* __restrict__ norm,
                          int E, int Etot) {
  long long idx = blockIdx.x * (long long)blockDim.x + threadIdx.x;
  int e = (int)(idx >> 5);
  if (e >= Etot) return;
  int q = (int)(idx & 31);
  int s, d;
  if (e < E) { s = ei[e]; d = ei[E + e]; } else { s = d = e - E; }
  float w = norm[e];
  float4 v = *((const float4*)(tmp + (size_t)s * D) + q);
  float* o = out + (size_t)d * D + (size_t)q * 4;
  atomicAdd(o + 0, v.x * w);
  atomicAdd(o + 1, v.y * w);
  atomicAdd(o + 2, v.z * w);
  atomicAdd(o + 3, v.w * w);
}

__global__ void k_relu(float* x, size_t total) {
  size_t i = blockIdx.x * (size_t)blockDim.x + threadIdx.x;
  if (i < total) x[i] = fmaxf(x[i], 0.0f);
}

// ---------------- fused link MLP: concat(h[s],h[d]) -> 128 relu -> 2 -> log_softmax ----
// One wave32 per 16-edge tile. Layer1 via WMMA f32 (K=256), z1 staged in padded LDS,
// layer2 (128->2) + log_softmax via VALU: 16 rows x 2 classes == 32 lanes.

__global__ void k_mlp(const float* __restrict__ h,
                      const int* __restrict__ e_pt, int n_pt,
                      const int* __restrict__ e_nt, int n_nt,
                      const int* __restrict__ e_ps, int n_ps,
                      const int* __restrict__ e_ns, int n_ns,
                      const float* __restrict__ Wl1, const float* __restrict__ bl1,
                      const float* __restrict__ Wl2, const float* __restrict__ bl2,
                      float* __restrict__ out) {
  __shared__ float z1[16][132];  // +4 words pad -> conflict-free transposed reads
  int lane = threadIdx.x;
  int l15  = lane & 15;
  int half = lane >> 4;
  int kb   = half * 2;
  int r = blockIdx.x * 16 + l15;  // global eval-edge row (grid sized so r always valid)

  const int* arr; int len, col;
  if (r < n_pt)                    { arr = e_pt; len = n_pt; col = r; }
  else if (r < n_pt + n_nt)        { arr = e_nt; len = n_nt; col = r - n_pt; }
  else if (r < n_pt + n_nt + n_ps) { arr = e_ps; len = n_ps; col = r - n_pt - n_nt; }
  else                             { arr = e_ns; len = n_ns; col = r - n_pt - n_nt - n_ps; }
  const float* rowS = h + (size_t)arr[col] * D;
  const float* rowD = h + (size_t)arr[len + col] * D;

  for (int nt = 0; nt < 8; ++nt) {
    int n = nt * 16 + l15;
    float bias = bl1[n];
    v8f c;
    #pragma unroll
    for (int v = 0; v < 8; ++v) c[v] = bias;
    for (int k0 = 0; k0 < 2 * D; k0 += 4) {
      int k = k0 + kb;                          // k even; k,k+1 on same side of 128
      const float* rsel = (k < D) ? rowS : rowD;
      int ko = (k < D) ? k : (k - D);
      v2f a, b;
      a[0] = rsel[ko];
      a[1] = rsel[ko + 1];
      b[0] = Wl1[(size_t)k * D + n];
      b[1] = Wl1[(size_t)(k + 1) * D + n];
      c = __builtin_amdgcn_wmma_f32_16x16x4_f32(false, a, false, b, (short)0, c,
                                                false, false);
    }
    #pragma unroll
    for (int v = 0; v < 8; ++v)
      z1[half * 8 + v][n] = fmaxf(c[v], 0.0f);
  }
  __syncthreads();  // single wave: LDS ops are in-order, this is effectively s_nop

  int rr  = lane >> 1;
  int cls = lane & 1;
  float acc = bl2[cls];
  #pragma unroll 8
  for (int k = 0; k < D; ++k)
    acc += z1[rr][k] * Wl2[k * 2 + cls];
  float other = __shfl_xor(acc, 1, 32);
  float mx  = fmaxf(acc, other);
  float lse = mx + logf(__expf(acc - mx) + __expf(other - mx));
  out[(size_t)(blockIdx.x * 16 + rr) * 2 + cls] = acc - lse;
}

// ---------------- launch ----------------

extern "C" void kernel_launch(void* const* d_in, const int* in_sizes, int n_in,
                              void* d_out, int out_size, void* d_ws, size_t ws_size,
                              hipStream_t stream) {
  const float* x   = (const float*)d_in[0];
  const float* W0  = (const float*)d_in[1];
  const float* b0  = (const float*)d_in[2];
  const float* W1  = (const float*)d_in[3];
  const float* b1  = (const float*)d_in[4];
  const float* W2  = (const float*)d_in[5];
  const float* b2  = (const float*)d_in[6];
  const float* W3  = (const float*)d_in[7];
  const float* b3  = (const float*)d_in[8];
  const float* Wl1 = (const float*)d_in[9];
  const float* bl1 = (const float*)d_in[10];
  const float* Wl2 = (const float*)d_in[11];
  const float* bl2 = (const float*)d_in[12];
  const int* ei   = (const int*)d_in[13];
  const int* e_pt = (const int*)d_in[14];
  const int* e_nt = (const int*)d_in[15];
  const int* e_ps = (const int*)d_in[16];
  const int* e_ns = (const int*)d_in[17];

  const int N    = in_sizes[0] / D;     // 100000 (divisible by 16)
  const int E    = in_sizes[13] / 2;    // 1,600,000
  const int Etot = E + N;               // + self loops
  const int n_pt = in_sizes[14] / 2, n_nt = in_sizes[15] / 2;
  const int n_ps = in_sizes[16] / 2, n_ns = in_sizes[17] / 2;
  const int R = n_pt + n_nt + n_ps + n_ns;  // 500000 (divisible by 16)

  char* ws = (char*)d_ws;
  size_t hbytes = (size_t)N * D * sizeof(float);
  float* hA   = (float*)(ws);
  float* hB   = (float*)(ws + hbytes);
  float* tmp  = (float*)(ws + 2 * hbytes);
  float* deg  = (float*)(ws + 3 * hbytes);
  float* dinv = (float*)(ws + 3 * hbytes + (size_t)N * sizeof(float));
  float* norm = (float*)(ws + 3 * hbytes + 2 * (size_t)N * sizeof(float));

  k_fill_deg <<<(N + 255) / 256, 256, 0, stream>>>(deg, N);
  k_count_deg<<<(E + 255) / 256, 256, 0, stream>>>(ei, deg, E);
  k_dinv     <<<(N + 255) / 256, 256, 0, stream>>>(deg, dinv, N);
  k_norm     <<<(Etot + 255) / 256, 256, 0, stream>>>(ei, dinv, norm, E, Etot);

  const int mtiles = N / 16;                       // 6250
  const int waves = mtiles * 8;                    // 50000 (exact, /4 waves per block)
  const int gemm_blocks = (waves + 3) / 4;
  const size_t tot = (size_t)N * D;
  const int elem_blocks = (int)((tot + 255) / 256);
  const long long scat_threads = (long long)Etot * 32;
  const int scat_blocks = (int)((scat_threads + 255) / 256);

  const float* Wls[4] = {W0, W1, W2, W3};
  const float* bls[4] = {b0, b1, b2, b3};
  float* outs[4] = {hA, hB, hA, hB};
  const float* in = x;
  for (int l = 0; l < 4; ++l) {
    float* o = outs[l];
    k_gemm128  <<<gemm_blocks, 128, 0, stream>>>(in, Wls[l], tmp, mtiles);
    k_init_bias<<<elem_blocks, 256, 0, stream>>>(o, bls[l], tot);
    k_scatter  <<<scat_blocks, 256, 0, stream>>>(tmp, o, ei, norm, E, Etot);
    if (l < 3) k_relu<<<elem_blocks, 256, 0, stream>>>(o, tot);
    in = o;
  }

  k_mlp<<<R / 16, 32, 0, stream>>>(hB, e_pt, n_pt, e_nt, n_nt, e_ps, n_ps, e_ns, n_ns,
                                   Wl1, bl1, Wl2, bl2, (float*)d_out);
}